// MergeHeads_88519275970643
// MI455X (gfx1250) — compile-verified
//
#include <hip/hip_runtime.h>

// ---------------------------------------------------------------------------
// Banked (MoE-style) linear:  y[t] = sum_a p[t,a] * (x[t,a] @ W[sel[t,a]] + b[sel[t,a]])
// T = 16384 tokens, A=2 slots, H=8 heads, DH=128, DM=2048.
//
// Tiling: one WG = 64 tokens x 64 DM cols, 4 wave32s (each wave: 16 cols,
// all 4 row-tiles -> 4 WMMA accumulators). K = H*DH = 1024 processed as 8
// head-chunks of 128; for each head the routed A tile (sel==h masked,
// prob-scaled) is built once cooperatively in LDS, then consumed by
// V_WMMA_F32_16X16X4_F32. Each B fragment (2 global loads) feeds 4 WMMAs.
// W traffic from L2 = (T/64)*K*DM*4B ~= 2.1 GB (4x less than 16-token tiles).
// ---------------------------------------------------------------------------

typedef float v2f __attribute__((ext_vector_type(2)));
typedef float v8f __attribute__((ext_vector_type(8)));

#define T_TOTAL 16384
#define A_SLOTS 2
#define H_HEADS 8
#define DH      128
#define DM      2048
#define TILE_T  64      // tokens per workgroup
#define BLOCK_N 64      // DM columns per workgroup (4 waves x 16 cols)
#define THREADS 128     // 4 wave32s
#define RT      4       // 16-row tiles per wave

__global__ __launch_bounds__(THREADS)
void banked_linear_wmma(const float* __restrict__ emb,     // [T, A, DH]
                        const int*   __restrict__ sel_idx, // [T, A]
                        const float* __restrict__ sel_p,   // [T, A]
                        const float* __restrict__ W,       // [H, DH, DM]
                        const float* __restrict__ bias,    // [H, DM]
                        float*       __restrict__ out)     // [T, DM]
{
    __shared__ float xw[TILE_T * A_SLOTS * DH];   // 64 KB prob-scaled inputs
    __shared__ float aT[TILE_T * DH];             // 32 KB routed A tile (one head)
    __shared__ float biasT[TILE_T * BLOCK_N];     // 16 KB weighted bias tile
    __shared__ int   selL[TILE_T * A_SLOTS];      // 512 B

    const int n0g = blockIdx.x * BLOCK_N;
    const int t0  = blockIdx.y * TILE_T;
    const int tid = threadIdx.x;

    // --- stage selected head indices -------------------------------------
    if (tid < TILE_T * A_SLOTS) selL[tid] = sel_idx[t0 * A_SLOTS + tid];

    // --- stage prob-scaled embeddings: 16384 floats as 4096 float4 -------
    {
        const float4* src = (const float4*)(emb + (size_t)t0 * A_SLOTS * DH);
        float4*       dst = (float4*)xw;
#pragma unroll
        for (int i = 0; i < 32; ++i) {
            int e        = tid + i * THREADS;     // float4 index
            int tok_slot = e >> 5;                // DH/4 = 32 float4 per (tok,slot)
            float p  = sel_p[t0 * A_SLOTS + tok_slot];
            float4 v = src[e];
            v.x *= p; v.y *= p; v.z *= p; v.w *= p;
            dst[e] = v;
        }
    }

    // --- weighted bias tile: biasT[r][j] = sum_a p*b[sel][n0g+j] ---------
#pragma unroll
    for (int i = 0; i < 32; ++i) {
        int e = tid + i * THREADS;                // 0..4095
        int r = e >> 6;                           // token row
        int j = e & 63;                           // column within tile
        int tt = (t0 + r) * A_SLOTS;
        float p0 = sel_p[tt + 0], p1 = sel_p[tt + 1];
        int   i0 = sel_idx[tt + 0], i1 = sel_idx[tt + 1];
        biasT[e] = p0 * bias[i0 * DM + n0g + j] + p1 * bias[i1 * DM + n0g + j];
    }

    __syncthreads();

    // --- per-wave setup ---------------------------------------------------
    const int lane = tid & 31;
    const int wave = tid >> 5;
    const int row  = lane & 15;          // A row within a 16-row tile
    const int hi   = lane >> 4;          // 0 -> K{0,1}, 1 -> K{2,3}
    const int koff = hi * 2;
    const int colB = lane & 15;          // B/C column owned by this lane
    const int n0   = n0g + wave * 16;

    v8f acc[RT] = {};                    // 4 x 16x16 f32 accumulators

    for (int h = 0; h < H_HEADS; ++h) {
        // cooperative routed-A build for head h: 8192 floats as 2048 float4
#pragma unroll
        for (int i = 0; i < 16; ++i) {
            int e  = tid + i * THREADS;          // float4 index
            int r  = e >> 5;                     // 32 float4 per row
            int d4 = e & 31;
            int s0 = selL[r * 2 + 0];
            int s1 = selL[r * 2 + 1];
            const float4* x0 = (const float4*)(xw + r * (A_SLOTS * DH)) + d4;
            const float4* x1 = x0 + (DH / 4);
            float4 a0 = *x0, a1 = *x1;
            float4 v;
            v.x = (s0 == h ? a0.x : 0.f) + (s1 == h ? a1.x : 0.f);
            v.y = (s0 == h ? a0.y : 0.f) + (s1 == h ? a1.y : 0.f);
            v.z = (s0 == h ? a0.z : 0.f) + (s1 == h ? a1.z : 0.f);
            v.w = (s0 == h ? a0.w : 0.f) + (s1 == h ? a1.w : 0.f);
            ((float4*)aT)[e] = v;
        }
        __syncthreads();

        const float* wp = W + (size_t)(h * DH) * DM + n0 + colB;
        __builtin_prefetch(wp, 0, 1);            // global_prefetch_b8 on W block
        const float* ap = aT + row * DH;

#pragma unroll 4
        for (int k4 = 0; k4 < DH; k4 += 4) {
            int kk = k4 + koff;                  // this lane's K pair
            // B fragment: W rows kk, kk+1 of head h, column n0+colB
            const float* wr = wp + (size_t)kk * DM;
            v2f b;
            b.x = wr[0];
            b.y = wr[DM];
            // 4 row-tiles share this B fragment -> 4 independent WMMA chains
#pragma unroll
            for (int rt = 0; rt < RT; ++rt) {
                v2f a = *(const v2f*)(ap + rt * 16 * DH + kk);
                acc[rt] = __builtin_amdgcn_wmma_f32_16x16x4_f32(
                    false, a, false, b, (short)0, acc[rt], false, false);
            }
        }
        __syncthreads();                         // aT reused next head
    }

    // --- bias add + store (C layout: VGPR v -> row v + 8*hi, lane&15 -> col)
#pragma unroll
    for (int rt = 0; rt < RT; ++rt) {
#pragma unroll
        for (int v = 0; v < 8; ++v) {
            int r = rt * 16 + v + hi * 8;
            float y = acc[rt][v] + biasT[r * BLOCK_N + wave * 16 + colB];
            out[(size_t)(t0 + r) * DM + n0 + colB] = y;
        }
    }
}

extern "C" void kernel_launch(void* const* d_in, const int* in_sizes, int n_in,
                              void* d_out, int out_size, void* d_ws, size_t ws_size,
                              hipStream_t stream) {
    (void)in_sizes; (void)n_in; (void)out_size; (void)d_ws; (void)ws_size;
    const float* emb  = (const float*)d_in[0];
    const int*   sidx = (const int*)d_in[1];
    const float* sp   = (const float*)d_in[2];
    const float* W    = (const float*)d_in[3];
    const float* b    = (const float*)d_in[4];
    float* out = (float*)d_out;

    dim3 grid(DM / BLOCK_N, T_TOTAL / TILE_T);   // (32, 256)
    banked_linear_wmma<<<grid, THREADS, 0, stream>>>(emb, sidx, sp, W, b, out);
}